// Model_38139309588622
// MI455X (gfx1250) — compile-verified
//
#include <hip/hip_runtime.h>

typedef __attribute__((ext_vector_type(2))) float v2f;
typedef __attribute__((ext_vector_type(4))) float v4f;
typedef __attribute__((ext_vector_type(8))) float v8f;

#define P_DIM 128

// ---------------------------------------------------------------------------
// Kernel 1: streaming reductions (memory-bound part, ~106 MB @ 23.3 TB/s).
//   - column sums of mu_N [n x 128]  -> ws_mu[block][128]
//   - pos/neg sums of h, hc, s, sc   -> ws_scal[block][8]
// One wave32 consumes one full 512B row per iteration (32 lanes x float4),
// perfectly coalesced, non-temporal (single-use stream).
// ---------------------------------------------------------------------------
__global__ __launch_bounds__(256) void reduce_kernel(
    const float* __restrict__ mu, const float* __restrict__ h,
    const float* __restrict__ hc, const float* __restrict__ s,
    const float* __restrict__ sc, int n,
    float* __restrict__ ws_mu, float* __restrict__ ws_scal)
{
    __shared__ float lds[2048];
    const int t    = threadIdx.x;
    const int lane = t & 31;   // wave32 on gfx1250
    const int wave = t >> 5;

    // ---- mu_N column sums: wave w handles rows gw, gw+gws, ... ----
    v4f acc = {0.f, 0.f, 0.f, 0.f};
    const int gw  = blockIdx.x * 8 + wave;
    const int gws = gridDim.x * 8;
    for (int row = gw; row < n; row += gws) {
        const v4f* p = reinterpret_cast<const v4f*>(mu + (size_t)row * P_DIM) + lane;
        v4f v = __builtin_nontemporal_load(p);
        acc += v;
    }
    lds[wave * 128 + lane * 4 + 0] = acc.x;
    lds[wave * 128 + lane * 4 + 1] = acc.y;
    lds[wave * 128 + lane * 4 + 2] = acc.z;
    lds[wave * 128 + lane * 4 + 3] = acc.w;
    __syncthreads();
    if (t < 128) {
        float sum = 0.f;
        #pragma unroll
        for (int w = 0; w < 8; ++w) sum += lds[w * 128 + t];
        ws_mu[(size_t)blockIdx.x * 128 + t] = sum;
    }
    __syncthreads();

    // ---- pos/neg sums of the four scalar streams (3% of the traffic) ----
    float ps[4] = {0.f, 0.f, 0.f, 0.f};
    float ns[4] = {0.f, 0.f, 0.f, 0.f};
    const float* arrs[4] = {h, hc, s, sc};
    const int gt  = blockIdx.x * 256 + t;
    const int gts = gridDim.x * 256;
    #pragma unroll
    for (int a = 0; a < 4; ++a) {
        const float* p = arrs[a];
        for (int i = gt; i < n; i += gts) {
            float v = __builtin_nontemporal_load(p + i);
            ps[a] += fmaxf(v, 0.f);
            ns[a] += fminf(v, 0.f);
        }
    }
    #pragma unroll
    for (int a = 0; a < 4; ++a) {
        lds[t * 8 + 2 * a + 0] = ps[a];
        lds[t * 8 + 2 * a + 1] = ns[a];
    }
    __syncthreads();
    if (t < 8) {
        float sum = 0.f;
        for (int i = 0; i < 256; ++i) sum += lds[i * 8 + t];
        ws_scal[(size_t)blockIdx.x * 8 + t] = sum;
    }
}

// ---------------------------------------------------------------------------
// Kernel 2: final reduce + five 128x128 f32 matvecs via V_WMMA_F32_16X16X4_F32
// + W10 @ xi + relu.  One block = 8 wave32; wave w owns output rows 16w..16w+15.
//
// WMMA f32 16x16x4 operand layout (per ISA 7.12.2), lane L, half h=L>>4:
//   A (16x4): m = L&15, VGPR v holds K = 2h + v  -> a = W[j0+m][k0+2h .. +1]
//   B (4x16): n = L&15, VGPR v holds K = 2h + v  -> b = vec[k0+2h .. +1] (bcast)
//   D (16x16): VGPR r holds D[r + 8h][L&15]      -> harvest column n==0
// ---------------------------------------------------------------------------
__global__ __launch_bounds__(256) void finalize_kernel(
    const float* __restrict__ ws_mu, const float* __restrict__ ws_scal, int nblk,
    const float* __restrict__ W1, const float* __restrict__ W2,
    const float* __restrict__ W4, const float* __restrict__ W6,
    const float* __restrict__ W8,
    const float* __restrict__ W3, const float* __restrict__ W5,
    const float* __restrict__ W7, const float* __restrict__ W9,
    const float* __restrict__ W10, const float* __restrict__ xi,
    float* __restrict__ out)
{
    __shared__ float vecs[5][128];
    __shared__ float scal[8];
    __shared__ float tmpv[128];
    const int t = threadIdx.x;

    // reduce per-block partials
    if (t < 128) {
        float sum = 0.f;
        for (int b = 0; b < nblk; ++b) sum += ws_mu[(size_t)b * 128 + t];
        vecs[0][t] = sum;
    }
    if (t >= 128 && t < 136) {
        const int j = t - 128;
        float sum = 0.f;
        for (int b = 0; b < nblk; ++b) sum += ws_scal[(size_t)b * 8 + j];
        scal[j] = sum;
    }
    __syncthreads();

    // branch(v, W)[k] = W[k] * (W[k] > 0 ? sum_pos(v) : sum_neg(v))
    if (t < 128) {
        float w;
        w = W3[t]; vecs[1][t] = w * (w > 0.f ? scal[0] : scal[1]);
        w = W5[t]; vecs[2][t] = w * (w > 0.f ? scal[2] : scal[3]);
        w = W7[t]; vecs[3][t] = w * (w > 0.f ? scal[4] : scal[5]);
        w = W9[t]; vecs[4][t] = w * (w > 0.f ? scal[6] : scal[7]);
    }
    __syncthreads();

    // five fused matvecs: tmp[j] = sum_mat sum_k vecs[mat][k] * Wmat[j][k]
    const float* Ws[5] = {W1, W2, W4, W6, W8};
    const int lane = t & 31;
    const int wave = t >> 5;
    const int j0   = wave * 16;
    const int m    = lane & 15;
    const int h2   = (lane >> 4) * 2;   // K offset of this lane-half

    v8f acc = {};
    for (int mat = 0; mat < 5; ++mat) {
        const float* wrow = Ws[mat] + (size_t)(j0 + m) * 128 + h2;
        const float* vrow = &vecs[mat][h2];
        #pragma unroll 4
        for (int k0 = 0; k0 < 128; k0 += 4) {
            v2f a, b;
            a.x = wrow[k0];
            a.y = wrow[k0 + 1];
            b.x = vrow[k0];
            b.y = vrow[k0 + 1];
            acc = __builtin_amdgcn_wmma_f32_16x16x4_f32(
                /*neg_a=*/false, a, /*neg_b=*/false, b,
                /*c_mod=*/(short)0, acc, /*reuse_a=*/false, /*reuse_b=*/false);
        }
    }
    // column n==0 lives in lanes 0 and 16; VGPR r = row (r + 8*half)
    if ((lane & 15) == 0) {
        const int mb = j0 + (lane >> 4) * 8;
        #pragma unroll
        for (int r = 0; r < 8; ++r) tmpv[mb + r] = acc[r];
    }
    __syncthreads();

    if (t < 128) {
        float v = tmpv[t] + W10[t * 3 + 0] * xi[0]
                          + W10[t * 3 + 1] * xi[1]
                          + W10[t * 3 + 2] * xi[2];
        out[t] = fmaxf(v, 0.f);
    }
}

extern "C" void kernel_launch(void* const* d_in, const int* in_sizes, int n_in,
                              void* d_out, int out_size, void* d_ws, size_t ws_size,
                              hipStream_t stream) {
    const float* xi   = (const float*)d_in[0];
    const float* mu_N = (const float*)d_in[1];
    const float* h    = (const float*)d_in[2];
    const float* hc   = (const float*)d_in[3];
    const float* s    = (const float*)d_in[4];
    const float* sc   = (const float*)d_in[5];
    const float* W1   = (const float*)d_in[6];
    const float* W2   = (const float*)d_in[7];
    const float* W3   = (const float*)d_in[8];
    const float* W4   = (const float*)d_in[9];
    const float* W5   = (const float*)d_in[10];
    const float* W6   = (const float*)d_in[11];
    const float* W7   = (const float*)d_in[12];
    const float* W8   = (const float*)d_in[13];
    const float* W9   = (const float*)d_in[14];
    const float* W10  = (const float*)d_in[15];
    float* out = (float*)d_out;

    const int n = in_sizes[2];  // N (h is (N,1))

    // per-block partials: 128 mu-column floats + 8 pos/neg scalars
    int nblk = (int)(ws_size / (136 * sizeof(float)));
    if (nblk > 512) nblk = 512;
    if (nblk < 1)   nblk = 1;

    float* ws_mu   = (float*)d_ws;
    float* ws_scal = ws_mu + (size_t)nblk * 128;

    reduce_kernel<<<nblk, 256, 0, stream>>>(mu_N, h, hc, s, sc, n, ws_mu, ws_scal);
    finalize_kernel<<<1, 256, 0, stream>>>(ws_mu, ws_scal, nblk,
                                           W1, W2, W4, W6, W8,
                                           W3, W5, W7, W9, W10, xi, out);
}